// densityDiffusionModule_47021301957210
// MI455X (gfx1250) — compile-verified
//
#include <hip/hip_runtime.h>

// ---------------------------------------------------------------------------
// delta-SPH density diffusion on MI455X (gfx1250).
// 3 edge passes + 2x2 pinv, segment sums via native global_atomic_add_f32.
// Inputs (setup_inputs order):
//   0: i        [E]   int32
//   1: j        [E]   int32
//   2: position [N,2] f32
//   3: volume   [N]   f32
//   4: distances(dir) [E,2] f32
//   5: radialDistances(q) [E] f32
//   6: density  [N]   f32
// Output: [N] f32
// ---------------------------------------------------------------------------

typedef float v2f __attribute__((ext_vector_type(2)));
typedef float v4f __attribute__((ext_vector_type(4)));

static constexpr float KGRAD  = -3.56507072e+05f;  // -20*(7/pi)/h^3, h=0.05
static constexpr float EPSQ   = 2.5e-04f;          // h^2 * 0.1
static constexpr float OUTC   = 1.21305396e-01f;   // SUPPORT*DELTA*C0
static constexpr float RESTD  = 1000.0f;
static constexpr float RCOND  = 2.3841858e-06f;    // 10*max(2,2)*eps_f32

// Device-scope, non-returning hardware f32 atomic add (L2-resolved, STOREcnt).
__device__ __forceinline__ void atomAddF(float* p, float v) {
    asm volatile("global_atomic_add_f32 %0, %1, off scope:SCOPE_DEV"
                 :: "v"(p), "v"(v) : "memory");
}

// Zero accumulators; must run every launch (harness poisons buffers).
__global__ void __launch_bounds__(256)
sph_init_kernel(v4f* __restrict__ M, v2f* __restrict__ gradRho,
                float* __restrict__ out, int N) {
    int n = blockIdx.x * blockDim.x + threadIdx.x;
    if (n < N) {
        M[n]       = (v4f){0.f, 0.f, 0.f, 0.f};
        gradRho[n] = (v2f){0.f, 0.f};
        out[n]     = 0.f;
    }
}

// Streamed (read-once) edge fetch with non-temporal hints.
__device__ __forceinline__ void edge_fetch(const int* __restrict__ I,
                                           const int* __restrict__ J,
                                           const v2f* __restrict__ dir,
                                           const float* __restrict__ rd,
                                           int e, int& pi, int& pj,
                                           float& gx, float& gy) {
    pi        = __builtin_nontemporal_load(I + e);
    pj        = __builtin_nontemporal_load(J + e);
    float q   = __builtin_nontemporal_load(rd + e);
    v2f   d   = __builtin_nontemporal_load(dir + e);
    float omq = 1.0f - q;
    float dw  = KGRAD * q * omq * omq * omq;   // dW/dr (Wendland C2, 2D)
    gx = d.x * dw;
    gy = d.y * dw;
}

// Pass A: M[i] += V_j * (r_j - r_i) (x) gradW
__global__ void __launch_bounds__(256)
sph_edgeA_kernel(const int* __restrict__ I, const int* __restrict__ J,
                 const v2f* __restrict__ pos, const float* __restrict__ vol,
                 const v2f* __restrict__ dir, const float* __restrict__ rd,
                 float* __restrict__ M, int E) {
    int e = blockIdx.x * blockDim.x + threadIdx.x;
    if (e >= E) return;
    int pi, pj; float gx, gy;
    edge_fetch(I, J, dir, rd, e, pi, pj, gx, gy);
    v2f xi = pos[pi], xj = pos[pj];
    float rbx = xj.x - xi.x, rby = xj.y - xi.y;
    float vj = vol[pj];
    float* Mi = M + 4 * pi;
    atomAddF(Mi + 0, vj * rbx * gx);
    atomAddF(Mi + 1, vj * rbx * gy);
    atomAddF(Mi + 2, vj * rby * gx);
    atomAddF(Mi + 3, vj * rby * gy);
}

// Particle pass: L = pinv(M), closed-form 2x2 Moore-Penrose (JAX rcond).
__global__ void __launch_bounds__(256)
sph_pinv_kernel(const v4f* __restrict__ M, v4f* __restrict__ L, int N) {
    int n = blockIdx.x * blockDim.x + threadIdx.x;
    if (n >= N) return;
    v4f m = M[n];
    float a = m.x, b = m.y, c = m.z, d = m.w;   // [[a,b],[c,d]]
    float Ee = 0.5f * (a + d), Ff = 0.5f * (a - d);
    float Gg = 0.5f * (c + b), Hh = 0.5f * (c - b);
    float Q  = sqrtf(Ee * Ee + Hh * Hh);
    float R  = sqrtf(Ff * Ff + Gg * Gg);
    float s1 = Q + R;                 // larger singular value
    float s2 = fabsf(Q - R);          // smaller singular value
    v4f r = (v4f){0.f, 0.f, 0.f, 0.f};
    if (s2 > RCOND * s1) {
        // full rank: pinv == inverse = adj(A)/det
        float det = a * d - b * c;
        float inv = 1.0f / det;
        r.x =  d * inv; r.y = -b * inv;
        r.z = -c * inv; r.w =  a * inv;
    } else if (s1 > 0.0f) {
        // rank-1 truncated pinv: A^T * P / s1^2,
        // P = (A A^T - s2^2 I) / (s1^2 - s2^2)  (dominant left-singular projector)
        float B00 = a * a + b * b;
        float B01 = a * c + b * d;
        float B11 = c * c + d * d;
        float s1sq = s1 * s1, s2sq = s2 * s2;
        float idn  = 1.0f / (s1sq - s2sq);
        float P00 = (B00 - s2sq) * idn;
        float P01 = B01 * idn;
        float P11 = (B11 - s2sq) * idn;
        float is1 = 1.0f / s1sq;
        // A^T = [[a,c],[b,d]]
        r.x = (a * P00 + c * P01) * is1;
        r.y = (a * P01 + c * P11) * is1;
        r.z = (b * P00 + d * P01) * is1;
        r.w = (b * P01 + d * P11) * is1;
    }
    L[n] = r;
}

// Pass B: gradRho[i] += -2 * rho_ba * V_j * (L[i] . gradW)
__global__ void __launch_bounds__(256)
sph_edgeB_kernel(const int* __restrict__ I, const int* __restrict__ J,
                 const float* __restrict__ vol,
                 const v2f* __restrict__ dir, const float* __restrict__ rd,
                 const float* __restrict__ den, const v4f* __restrict__ L,
                 float* __restrict__ gradRho, int E) {
    int e = blockIdx.x * blockDim.x + threadIdx.x;
    if (e >= E) return;
    int pi, pj; float gx, gy;
    edge_fetch(I, J, dir, rd, e, pi, pj, gx, gy);
    v4f Li = L[pi];
    float gradx = Li.x * gx + Li.y * gy;
    float grady = Li.z * gx + Li.w * gy;
    float rho_ba = RESTD * den[pj] - RESTD * den[pi];
    float s = -2.0f * rho_ba * vol[pj];
    atomAddF(gradRho + 2 * pi + 0, s * gradx);
    atomAddF(gradRho + 2 * pi + 1, s * grady);
}

// Pass C: out[i] += OUTC * V_j * dot(psi, gradW)
__global__ void __launch_bounds__(256)
sph_edgeC_kernel(const int* __restrict__ I, const int* __restrict__ J,
                 const v2f* __restrict__ pos, const float* __restrict__ vol,
                 const v2f* __restrict__ dir, const float* __restrict__ rd,
                 const float* __restrict__ den, const v2f* __restrict__ gradRho,
                 float* __restrict__ out, int E) {
    int e = blockIdx.x * blockDim.x + threadIdx.x;
    if (e >= E) return;
    int pi, pj; float gx, gy;
    edge_fetch(I, J, dir, rd, e, pi, pj, gx, gy);
    v2f xi = pos[pi], xj = pos[pj];
    float rbx = xj.x - xi.x, rby = xj.y - xi.y;
    float rji2 = rbx * rbx + rby * rby + EPSQ;
    float rho_ba = RESTD * den[pj] - RESTD * den[pi];
    float coef = 2.0f * rho_ba / rji2;
    v2f gi = gradRho[pi], gj = gradRho[pj];
    float psix = coef * rbx - (gi.x + gj.x);
    float psiy = coef * rby - (gi.y + gj.y);
    float prod = psix * gx + psiy * gy;
    atomAddF(out + pi, OUTC * prod * vol[pj]);
}

extern "C" void kernel_launch(void* const* d_in, const int* in_sizes, int n_in,
                              void* d_out, int out_size, void* d_ws, size_t ws_size,
                              hipStream_t stream) {
    const int*   I   = (const int*)  d_in[0];
    const int*   J   = (const int*)  d_in[1];
    const v2f*   pos = (const v2f*)  d_in[2];
    const float* vol = (const float*)d_in[3];
    const v2f*   dir = (const v2f*)  d_in[4];
    const float* rd  = (const float*)d_in[5];
    const float* den = (const float*)d_in[6];
    float* out = (float*)d_out;

    const int E = in_sizes[0];   // edges
    const int N = in_sizes[3];   // particles

    // Workspace layout (16B aligned): M [4N f32] | L [4N f32] | gradRho [2N f32]
    char* ws = (char*)d_ws;
    float* M       = (float*)(ws);
    float* L       = (float*)(ws + (size_t)16 * N);
    float* gradRho = (float*)(ws + (size_t)32 * N);

    const int B = 256;
    const int gN = (N + B - 1) / B;
    const int gE = (E + B - 1) / B;

    sph_init_kernel <<<gN, B, 0, stream>>>((v4f*)M, (v2f*)gradRho, out, N);
    sph_edgeA_kernel<<<gE, B, 0, stream>>>(I, J, pos, vol, dir, rd, M, E);
    sph_pinv_kernel <<<gN, B, 0, stream>>>((const v4f*)M, (v4f*)L, N);
    sph_edgeB_kernel<<<gE, B, 0, stream>>>(I, J, vol, dir, rd, den, (const v4f*)L,
                                           gradRho, E);
    sph_edgeC_kernel<<<gE, B, 0, stream>>>(I, J, pos, vol, dir, rd, den,
                                           (const v2f*)gradRho, out, E);
}